// ParaphraseVAE_77764677861451
// MI455X (gfx1250) — compile-verified
//
#include <hip/hip_runtime.h>
#include <hip/hip_bf16.h>
#include <stdint.h>

// ---------------------------------------------------------------------------
// ParaphraseVAE on gfx1250 (MI455X).
// - All dense matrix math via V_WMMA_F32_16X16X4_F32 (fp32-exact, wave32).
// - Output projection [60,600]x[600,50257] streams out_W (120MB) exactly once,
//   staged per 16-row tile into LDS by the Tensor Data Mover
//   (tensor_load_to_lds + s_wait_tensorcnt), HBM-bound ~5us @ 23.3 TB/s.
// - Sequential GRU/attention chains run as single-workgroup kernels with
//   LDS-resident hidden state; encoder input gates are hoisted into a WMMA
//   GEMM since they don't depend on the recurrent state.
// ---------------------------------------------------------------------------

typedef __attribute__((ext_vector_type(2))) float v2f;
typedef __attribute__((ext_vector_type(8))) float v8f;
typedef __attribute__((ext_vector_type(4))) unsigned int u32x4;
typedef __attribute__((ext_vector_type(8))) int i32x8;
typedef __attribute__((ext_vector_type(4))) int i32x4;

#define VOCAB   50257
#define EDIM    300
#define HDIM    600
#define HD      300
#define LDIM    60
#define ZDIM    100
#define T_IN    48
#define T_OUT   60
#define DEC_ELEMS ((size_t)T_OUT * VOCAB)

__device__ __forceinline__ v8f wmma4(v2f a, v2f b, v8f c) {
  return __builtin_amdgcn_wmma_f32_16x16x4_f32(false, a, false, b, (short)0, c,
                                               false, false);
}

// ---------------------------------------------------------------------------
// Generic small GEMM: C[M,N] = A[M,K] @ W[N,K]^T + bias.  One wave per 16x16
// tile; fp32 WMMA over K in steps of 4 (all our K are multiples of 4).
// Boundary handling: clamp load rows for address safety only — OOB A rows /
// B cols can only pollute C rows/cols that the guarded store never writes.
// ---------------------------------------------------------------------------
__global__ __launch_bounds__(32) void gemm_bias_wmma(
    const float* __restrict__ A, int lda, int M,
    const float* __restrict__ W, int ldw, int N, int K,
    const float* __restrict__ bias, float* __restrict__ C, int ldc) {
  int n0 = blockIdx.x * 16, m0 = blockIdx.y * 16;
  int lane = threadIdx.x;
  int half = lane >> 4, i16 = lane & 15;
  int m = m0 + i16, n = n0 + i16;
  const float* Arow = A + (size_t)((m < M) ? m : (M - 1)) * lda + 2 * half;
  const float* Wrow = W + (size_t)((n < N) ? n : (N - 1)) * ldw + 2 * half;
  v8f acc = {};
  for (int k0 = 0; k0 < K; k0 += 4) {
    v2f a, b;
    a.x = Arow[k0];
    a.y = Arow[k0 + 1];
    b.x = Wrow[k0];
    b.y = Wrow[k0 + 1];
    acc = wmma4(a, b, acc);
  }
  if (n < N) {
    float bv = bias[n];
#pragma unroll
    for (int v = 0; v < 8; ++v) {
      int mm = m0 + v + 8 * half;
      if (mm < M) C[(size_t)mm * ldc + n] = acc[v] + bv;
    }
  }
}

// ---------------------------------------------------------------------------
// Output projection: out[60,VOCAB] = H[60,600] @ out_W[VOCAB,600]^T + out_b.
// Block = 4 waves; block owns one 16-column (vocab) tile. The 16 rows of
// out_W (row-major [V,600]) form one contiguous 2-D tile that the Tensor
// Data Mover DMAs into LDS (tile 600x16, stride 600); tensor_dim1 = N-n0
// gives hardware zero-fill for the ragged last tile. Each wave then computes
// one 16-row M-tile against the shared B tile.
// ---------------------------------------------------------------------------
__global__ __launch_bounds__(128) void out_proj_wmma(
    const float* __restrict__ H, const float* __restrict__ W,
    const float* __restrict__ bias, float* __restrict__ out, int N) {
  __shared__ float sB[16 * HDIM];
  int n0 = blockIdx.x * 16;

  if (threadIdx.x < 32) {
    // Build the TDM descriptor (D#) for a 2-D tile load: 600 floats x 16 rows.
    uint32_t lds = (uint32_t)(uintptr_t)(&sB[0]);
    uint64_t ga = (uint64_t)(uintptr_t)W + (uint64_t)n0 * (HDIM * 4u);
    uint32_t tdim1 = (uint32_t)(N - n0);  // valid rows; OOB rows read as zero
    u32x4 g0 = {1u,                              // count=1, user descriptor
                lds,                             // lds_addr (bytes)
                (uint32_t)ga,                    // global_addr[31:0]
                (uint32_t)((ga >> 32) & 0x01FFFFFFu) | 0x80000000u};  // type=2
    i32x8 g1 = {(int)0x00020000u,                  // data_size=4B
                (int)(600u << 16),                 // tensor_dim0 = 600
                (int)((tdim1 & 0xFFFFu) << 16),    // tensor_dim1 lo16
                (int)(((tdim1 >> 16) & 0xFFFFu) | (600u << 16)),  // tile_dim0
                16,                                // tile_dim1 = 16 rows
                600,                               // tensor_dim0_stride = 600
                0, 0};
    i32x4 z4 = {0, 0, 0, 0};
#if defined(__clang_major__) && __clang_major__ >= 23
    i32x8 z8 = {0, 0, 0, 0, 0, 0, 0, 0};
    __builtin_amdgcn_tensor_load_to_lds(g0, g1, z4, z4, z8, 0);
#else
    __builtin_amdgcn_tensor_load_to_lds(g0, g1, z4, z4, 0);
#endif
    __builtin_amdgcn_s_wait_tensorcnt(0);
  }
  __syncthreads();

  int wave = threadIdx.x >> 5;
  int lane = threadIdx.x & 31;
  int half = lane >> 4, i16 = lane & 15;
  int m0 = wave * 16;
  int m = m0 + i16;
  const float* Arow = H + (size_t)((m < T_OUT) ? m : 0) * HDIM + 2 * half;
  const float* Brow = sB + i16 * HDIM + 2 * half;
  v8f acc = {};
#pragma unroll 5
  for (int k0 = 0; k0 < HDIM; k0 += 4) {
    v2f a, b;
    a.x = Arow[k0];
    a.y = Arow[k0 + 1];
    b.x = Brow[k0];
    b.y = Brow[k0 + 1];
    acc = wmma4(a, b, acc);
  }
  int n = n0 + i16;
  if (n < N) {
    float bv = bias[n];
#pragma unroll
    for (int v = 0; v < 8; ++v) {
      int mm = m0 + v + 8 * half;
      if (mm < T_OUT) out[(size_t)mm * N + n] = acc[v] + bv;
    }
  }
}

// ---------------------------------------------------------------------------
// Gather embedding rows for the encoder input sequence.
// ---------------------------------------------------------------------------
__global__ void gather_emb(const int* __restrict__ toks,
                           const float* __restrict__ emb,
                           float* __restrict__ X) {
  int t = blockIdx.x;
  int tok = toks[t];
  for (int i = threadIdx.x; i < EDIM; i += blockDim.x)
    X[t * EDIM + i] = emb[(size_t)tok * EDIM + i];
}

// ---------------------------------------------------------------------------
// Encoder recurrence. Input gates Gi = Wih@x + bih are precomputed by WMMA;
// only Whh@h remains in the sequential chain. Hidden state lives in LDS.
// ---------------------------------------------------------------------------
__global__ __launch_bounds__(1024) void enc_recurrence(
    const float* __restrict__ Gi_f, const float* __restrict__ Gi_b,
    const float* __restrict__ Whh_f, const float* __restrict__ Whh_b,
    const float* __restrict__ bhh_f, const float* __restrict__ bhh_b,
    float* __restrict__ enc_out, float* __restrict__ hfinal) {
  __shared__ float sh[2 * HD];
  __shared__ float sgh[6 * HD];  // 1800 gate pre-activations
  int t = threadIdx.x;
  for (int i = t; i < 2 * HD; i += 1024) sh[i] = 0.0f;
  __syncthreads();
  for (int step = 0; step < T_IN; ++step) {
    for (int g = t; g < 1800; g += 1024) {
      int dir = g / 900, i = g % 900;
      const float* Wrow = (dir ? Whh_b : Whh_f) + i * HD;
      const float* hh = sh + dir * HD;
      float s = (dir ? bhh_b : bhh_f)[i];
      for (int k = 0; k < HD; ++k) s += Wrow[k] * hh[k];
      sgh[g] = s;
    }
    __syncthreads();
    if (t < 2 * HD) {
      int dir = t / HD, i = t % HD;
      const float* gi = (dir ? Gi_b : Gi_f) + step * 900;
      const float* gh = sgh + dir * 900;
      float r = 1.0f / (1.0f + __expf(-(gi[i] + gh[i])));
      float zg = 1.0f / (1.0f + __expf(-(gi[HD + i] + gh[HD + i])));
      float nn = tanhf(gi[2 * HD + i] + r * gh[2 * HD + i]);
      float hn = (1.0f - zg) * nn + zg * sh[t];
      sh[t] = hn;
      enc_out[step * HDIM + t] = hn;
    }
    __syncthreads();
  }
  for (int i = t; i < 2 * HD; i += 1024) hfinal[i] = sh[i];
  for (int i = t; i < (LDIM - T_IN) * HDIM; i += 1024)
    enc_out[T_IN * HDIM + i] = 0.0f;  // zero-pad rows 48..59
}

// ---------------------------------------------------------------------------
// Reparameterization: z = eps * exp(0.5*logv) + mean  (elementwise, [60,100])
// ---------------------------------------------------------------------------
__global__ void reparam(const float* __restrict__ eps,
                        const float* __restrict__ mean,
                        const float* __restrict__ logv,
                        float* __restrict__ z, int n) {
  int i = blockIdx.x * blockDim.x + threadIdx.x;
  if (i < n) z[i] = eps[i] * __expf(0.5f * logv[i]) + mean[i];
}

// ---------------------------------------------------------------------------
// Decoder recurrence (teacher forcing). Attention + comb + bidirectional GRU
// per step, all state in LDS; emits H states to global for the batched
// output projection.
// ---------------------------------------------------------------------------
__global__ __launch_bounds__(1024) void dec_recurrence(
    const int* __restrict__ target, const float* __restrict__ emb,
    const float* __restrict__ latent_out,
    const float* __restrict__ attn_W, const float* __restrict__ attn_b,
    const float* __restrict__ comb_W, const float* __restrict__ comb_b,
    const float* __restrict__ Wih_f, const float* __restrict__ Whh_f,
    const float* __restrict__ bih_f, const float* __restrict__ bhh_f,
    const float* __restrict__ Wih_b, const float* __restrict__ Whh_b,
    const float* __restrict__ bih_b, const float* __restrict__ bhh_b,
    const float* __restrict__ hinit, float* __restrict__ decH) {
  __shared__ float sx[EDIM];
  __shared__ float sh[HDIM];
  __shared__ float sap[HDIM];
  __shared__ float so[EDIM];
  __shared__ float sgi[1800];
  __shared__ float sgh[1800];
  __shared__ float spart[960];
  __shared__ float saw[LDIM];
  __shared__ float sred[2];
  int t = threadIdx.x;
  for (int i = t; i < HDIM; i += 1024) sh[i] = hinit[i];
  __syncthreads();
  for (int step = 0; step < T_OUT; ++step) {
    int tok = (step == 0) ? 1 : target[step - 1];  // EOS prepended
    for (int i = t; i < EDIM; i += 1024) sx[i] = emb[(size_t)tok * EDIM + i];
    __syncthreads();
    // attention scores: attn_W @ [x, hf, hb] + attn_b, split-16 per row
    if (t < 960) {
      int j = t >> 4, p = t & 15;
      const float* w = attn_W + j * 900;
      float s = 0.0f;
      for (int k = p; k < 900; k += 16) {
        float in = (k < EDIM) ? sx[k] : sh[k - EDIM];
        s += w[k] * in;
      }
      spart[t] = s;
    }
    __syncthreads();
    if (t < LDIM) {
      float s = attn_b[t];
      for (int p = 0; p < 16; ++p) s += spart[t * 16 + p];
      saw[t] = s;
    }
    __syncthreads();
    if (t == 0) {
      float mx = saw[0];
      for (int j = 1; j < LDIM; ++j) mx = fmaxf(mx, saw[j]);
      sred[0] = mx;
    }
    __syncthreads();
    if (t < LDIM) saw[t] = __expf(saw[t] - sred[0]);
    __syncthreads();
    if (t == 0) {
      float s = 0.0f;
      for (int j = 0; j < LDIM; ++j) s += saw[j];
      sred[1] = 1.0f / s;
    }
    __syncthreads();
    if (t < LDIM) saw[t] *= sred[1];
    __syncthreads();
    // applied = aw @ latent_out   [600]
    if (t < HDIM) {
      float s = 0.0f;
      for (int j = 0; j < LDIM; ++j) s += saw[j] * latent_out[j * HDIM + t];
      sap[t] = s;
    }
    __syncthreads();
    // o = relu(comb_W @ [x, applied] + comb_b), split-3 per row
    if (t < 900) {
      int i = t % EDIM, p = t / EDIM;
      const float* w = comb_W + i * 900 + p * 300;
      float s = 0.0f;
      for (int k = 0; k < 300; ++k) {
        int kk = p * 300 + k;
        float in = (kk < EDIM) ? sx[kk] : sap[kk - EDIM];
        s += w[k] * in;
      }
      spart[t] = s;
    }
    __syncthreads();
    if (t < EDIM) {
      float s = comb_b[t] + spart[t] + spart[EDIM + t] + spart[2 * EDIM + t];
      so[t] = fmaxf(s, 0.0f);
    }
    __syncthreads();
    // bidirectional GRU gates
    for (int g = t; g < 1800; g += 1024) {
      int dir = g / 900, i = g % 900;
      const float* wi = (dir ? Wih_b : Wih_f) + i * HD;
      const float* wh = (dir ? Whh_b : Whh_f) + i * HD;
      const float* hh = sh + dir * HD;
      float si = (dir ? bih_b : bih_f)[i];
      float sg = (dir ? bhh_b : bhh_f)[i];
      for (int k = 0; k < HD; ++k) {
        si += wi[k] * so[k];
        sg += wh[k] * hh[k];
      }
      sgi[g] = si;
      sgh[g] = sg;
    }
    __syncthreads();
    if (t < HDIM) {
      int dir = t / HD, i = t % HD;
      const float* gi = sgi + dir * 900;
      const float* gh = sgh + dir * 900;
      float r = 1.0f / (1.0f + __expf(-(gi[i] + gh[i])));
      float zg = 1.0f / (1.0f + __expf(-(gi[HD + i] + gh[HD + i])));
      float nn = tanhf(gi[2 * HD + i] + r * gh[2 * HD + i]);
      float hn = (1.0f - zg) * nn + zg * sh[t];
      sh[t] = hn;
      decH[step * HDIM + t] = hn;
    }
    __syncthreads();
  }
}

// ---------------------------------------------------------------------------
// Row-wise log-softmax over the vocab dim, in place. One block per row.
// ---------------------------------------------------------------------------
__global__ __launch_bounds__(256) void log_softmax_rows(float* __restrict__ p,
                                                        int N) {
  __shared__ float red[256];
  float* row = p + (size_t)blockIdx.x * N;
  int t = threadIdx.x;
  float mx = -3.402823466e38f;
  for (int i = t; i < N; i += 256) mx = fmaxf(mx, row[i]);
  red[t] = mx;
  __syncthreads();
  for (int s = 128; s > 0; s >>= 1) {
    if (t < s) red[t] = fmaxf(red[t], red[t + s]);
    __syncthreads();
  }
  mx = red[0];
  __syncthreads();
  float sum = 0.0f;
  for (int i = t; i < N; i += 256) sum += __expf(row[i] - mx);
  red[t] = sum;
  __syncthreads();
  for (int s = 128; s > 0; s >>= 1) {
    if (t < s) red[t] += red[t + s];
    __syncthreads();
  }
  float lse = mx + __logf(red[0]);
  for (int i = t; i < N; i += 256) row[i] -= lse;
}

// ---------------------------------------------------------------------------
extern "C" void kernel_launch(void* const* d_in, const int* in_sizes, int n_in,
                              void* d_out, int out_size, void* d_ws,
                              size_t ws_size, hipStream_t stream) {
  const int*   input    = (const int*)d_in[0];
  const int*   target   = (const int*)d_in[1];
  const float* eps      = (const float*)d_in[2];
  const float* emb      = (const float*)d_in[3];
  const float* eWih_f   = (const float*)d_in[4];
  const float* eWhh_f   = (const float*)d_in[5];
  const float* ebih_f   = (const float*)d_in[6];
  const float* ebhh_f   = (const float*)d_in[7];
  const float* eWih_b   = (const float*)d_in[8];
  const float* eWhh_b   = (const float*)d_in[9];
  const float* ebih_b   = (const float*)d_in[10];
  const float* ebhh_b   = (const float*)d_in[11];
  const float* dWih_f   = (const float*)d_in[12];
  const float* dWhh_f   = (const float*)d_in[13];
  const float* dbih_f   = (const float*)d_in[14];
  const float* dbhh_f   = (const float*)d_in[15];
  const float* dWih_b   = (const float*)d_in[16];
  const float* dWhh_b   = (const float*)d_in[17];
  const float* dbih_b   = (const float*)d_in[18];
  const float* dbhh_b   = (const float*)d_in[19];
  const float* h2m_W    = (const float*)d_in[20];
  const float* h2m_b    = (const float*)d_in[21];
  const float* h2v_W    = (const float*)d_in[22];
  const float* h2v_b    = (const float*)d_in[23];
  const float* l2h_W    = (const float*)d_in[24];
  const float* l2h_b    = (const float*)d_in[25];
  const float* attn_W   = (const float*)d_in[26];
  const float* attn_b   = (const float*)d_in[27];
  const float* comb_W   = (const float*)d_in[28];
  const float* comb_b   = (const float*)d_in[29];
  const float* out_W    = (const float*)d_in[30];
  const float* out_b    = (const float*)d_in[31];

  float* out    = (float*)d_out;
  float* mean   = out + DEC_ELEMS;                // [60,100]
  float* logv   = out + DEC_ELEMS + LDIM * ZDIM;  // [60,100]

  // workspace layout (floats)
  float* ws       = (float*)d_ws;
  float* X_in     = ws;                 // 48*300   = 14400
  float* Gi_f     = X_in + 14400;       // 48*900   = 43200
  float* Gi_b     = Gi_f + 43200;       // 43200
  float* enc_out  = Gi_b + 43200;       // 60*600   = 36000
  float* hfinal   = enc_out + 36000;    // 600
  float* zbuf     = hfinal + 600;       // 6000
  float* latent   = zbuf + 6000;        // 60*600   = 36000
  float* decH     = latent + 36000;     // 60*600   = 36000

  // 1) gather encoder input embeddings
  gather_emb<<<dim3(T_IN), dim3(256), 0, stream>>>(input, emb, X_in);

  // 2) batched encoder input gates via WMMA: Gi = X @ Wih^T + bih  [48,900]
  gemm_bias_wmma<<<dim3(57, 3), dim3(32), 0, stream>>>(
      X_in, EDIM, T_IN, eWih_f, EDIM, 900, EDIM, ebih_f, Gi_f, 900);
  gemm_bias_wmma<<<dim3(57, 3), dim3(32), 0, stream>>>(
      X_in, EDIM, T_IN, eWih_b, EDIM, 900, EDIM, ebih_b, Gi_b, 900);

  // 3) sequential encoder recurrence
  enc_recurrence<<<dim3(1), dim3(1024), 0, stream>>>(
      Gi_f, Gi_b, eWhh_f, eWhh_b, ebhh_f, ebhh_b, enc_out, hfinal);

  // 4) VAE heads via WMMA: mean/logv = enc_out @ W^T + b  [60,100]
  gemm_bias_wmma<<<dim3(7, 4), dim3(32), 0, stream>>>(
      enc_out, HDIM, LDIM, h2m_W, HDIM, ZDIM, HDIM, h2m_b, mean, ZDIM);
  gemm_bias_wmma<<<dim3(7, 4), dim3(32), 0, stream>>>(
      enc_out, HDIM, LDIM, h2v_W, HDIM, ZDIM, HDIM, h2v_b, logv, ZDIM);

  // 5) reparameterize
  reparam<<<dim3((LDIM * ZDIM + 255) / 256), dim3(256), 0, stream>>>(
      eps, mean, logv, zbuf, LDIM * ZDIM);

  // 6) latent_out = z @ l2h_W^T + b  [60,600] via WMMA
  gemm_bias_wmma<<<dim3(38, 4), dim3(32), 0, stream>>>(
      zbuf, ZDIM, LDIM, l2h_W, ZDIM, HDIM, ZDIM, l2h_b, latent, HDIM);

  // 7) sequential decoder recurrence (stores all 60 hidden states)
  dec_recurrence<<<dim3(1), dim3(1024), 0, stream>>>(
      target, emb, latent, attn_W, attn_b, comb_W, comb_b,
      dWih_f, dWhh_f, dbih_f, dbhh_f, dWih_b, dWhh_b, dbih_b, dbhh_b,
      hfinal, decH);

  // 8) batched output projection (TDM-staged B tiles, fp32 WMMA, HBM-bound)
  out_proj_wmma<<<dim3((VOCAB + 15) / 16), dim3(128), 0, stream>>>(
      decH, out_W, out_b, out, VOCAB);

  // 9) in-place log-softmax per output row
  log_softmax_rows<<<dim3(T_OUT), dim3(256), 0, stream>>>(out, VOCAB);
}